// QuantizeEma_27161373180474
// MI455X (gfx1250) — compile-verified
//
#include <hip/hip_runtime.h>

// ---------------------------------------------------------------------------
// VQ-VAE EMA codebook update for MI455X (gfx1250, wave32, WMMA + TDM).
// Heavy phase: fp32 GEMM-argmin via V_WMMA_F32_16X16X4_F32 with codebook
// tiles DMA'd into double-buffered LDS by the Tensor Data Mover
// (tensor_load_to_lds / s_wait_tensorcnt), two independent accumulator chains.
// ---------------------------------------------------------------------------

#define K_EMBED 10000
#define E_DIM   128
#define N_SAMP  16384           // 16*32*32
#define DECAYF  0.99f
#define ONEMD   0.01f
#define EPSF    1e-5f

typedef __attribute__((ext_vector_type(2))) float        v2f;
typedef __attribute__((ext_vector_type(8))) float        v8f;
typedef __attribute__((ext_vector_type(4))) unsigned int v4u;
typedef __attribute__((ext_vector_type(4))) int          v4i;
typedef __attribute__((ext_vector_type(8))) int          v8i;

#if __has_builtin(__builtin_amdgcn_tensor_load_to_lds) && \
    __has_builtin(__builtin_amdgcn_s_wait_tensorcnt)
#define USE_TDM 1
#else
#define USE_TDM 0
#endif

// ---- workspace layout (float offsets) ----
#define WS_CNORM   0            // 10000
#define WS_IDX     10240        // 16384 ints
#define WS_COUNTS  26880        // 10000
#define WS_SUMS    36880        // 1,280,000
#define WS_NS      1316880      // 1
#define WS_ZERO_N  (WS_NS - WS_COUNTS + 1)   // counts..nsample contiguous

// ---- output layout (float offsets into d_out) ----
#define OUT_Q      0            // 2,097,152
#define OUT_LOSS   2097152      // 1
#define OUT_MEAN   2097153      // 1,280,000
#define OUT_SIZE   3377153      // 10,000
#define OUT_SUM    3387153      // 1,280,000

// ---------------------------------------------------------------------------
__global__ void k_colnorm(const float* __restrict__ cm, float* __restrict__ cnorm) {
    int k = blockIdx.x * 256 + threadIdx.x;
    if (k >= K_EMBED) return;
    float s = 0.0f;
    #pragma unroll 8
    for (int e = 0; e < E_DIM; ++e) {
        float m = cm[e * K_EMBED + k];
        s = fmaf(m, m, s);
    }
    cnorm[k] = s;
}

// ---------------------------------------------------------------------------
__global__ void k_zero(float* __restrict__ z, int n, float* __restrict__ loss) {
    int i = blockIdx.x * 256 + threadIdx.x;
    if (i < n) z[i] = 0.0f;
    if (i == 0) *loss = 0.0f;
}

// ---------------------------------------------------------------------------
#if USE_TDM
// Issue a TDM load of a 128x16 f32 codebook tile (rows = embed dims, 16
// contiguous codes per row, row stride K_EMBED) into LDS at lds_off.
// D# per CDNA5 ISA: group0 = {flags, lds_addr, gaddr57,
// type=2}; group1 = {data_size=4B, tensor_dim0/1, tile_dim0/1, stride0};
// groups 2/3 (and the trailing group on the 6-arg toolchain) zero: 2-D tensor.
__device__ __forceinline__ void tdm_load_tile(const float* gsrc, unsigned int lds_off) {
    unsigned long long ga = (unsigned long long)(uintptr_t)gsrc;
    v4u g0;
    g0.x = 1u;                                   // count=1, is_restore=0, no gather
    g0.y = lds_off;                              // LDS destination (bytes)
    g0.z = (unsigned int)ga;                     // global_addr[31:0]
    g0.w = (unsigned int)(ga >> 32) | (2u << 30);// global_addr[56:32] | type=2
    v8i g1;
    g1[0] = 0x20000;                             // wg_mask=0 | data_size=2 (4B)
    g1[1] = (K_EMBED << 16);                     // tensor_dim0[15:0] @ [31:16]
    g1[2] = (E_DIM << 16);                       // tensor_dim1[15:0] @ [31:16]
    g1[3] = (16 << 16);                          // tile_dim0=16 @ [31:16]
    g1[4] = E_DIM;                               // tile_dim1=128, tile_dim2=0
    g1[5] = K_EMBED;                             // tensor_dim0_stride[31:0]
    g1[6] = 0;
    g1[7] = 0;
    v4i gz4 = {0, 0, 0, 0};
    v8i gz8 = {0, 0, 0, 0, 0, 0, 0, 0};
    __builtin_amdgcn_tensor_load_to_lds(g0, g1, gz4, gz4, gz8, 0);
}
#endif

// ---------------------------------------------------------------------------
// GEMM-argmin. Per wave: 16 sample rows x all 10000 codes.
// Block = 8 waves = 128 rows; grid = 128 blocks = 16384 rows.
__global__ __launch_bounds__(256) void k_argmin(const float* __restrict__ x,
                                                const float* __restrict__ cm,
                                                const float* __restrict__ cnorm,
                                                int* __restrict__ idx) {
#if USE_TDM
    __shared__ float lds_b[2][E_DIM * 16];  // double-buffered, row-major e*16+k
#else
    __shared__ float lds_b[1][E_DIM * 16];  // paired layout p*32 + k*2 + (e&1)
#endif

    const int lane = threadIdx.x & 31;
    const int wave = threadIdx.x >> 5;
    const int half = lane >> 4;             // K-pair selector (A/B layout)
    const int l16  = lane & 15;
    const int rowBase = blockIdx.x * 128 + wave * 16;

    // Preload A fragments: 32 chunks of K=4 over E=128.
    // 16x4 fp32 A layout: lanes 0-15 -> M, v0/v1 = K pair (half selects K-pair).
    v2f a[32];
    const float* xrow = x + (size_t)(rowBase + l16) * E_DIM;
    #pragma unroll
    for (int i = 0; i < 32; ++i) {
        int e = 4 * i + 2 * half;
        a[i] = *(const v2f*)(xrow + e);
    }

    float best_v[8];
    int   best_i[8];
    #pragma unroll
    for (int j = 0; j < 8; ++j) { best_v[j] = 3.4e38f; best_i[j] = 0; }

#if USE_TDM
    const bool issuer = (wave == 0);
    const unsigned int lds0 = (unsigned int)(uintptr_t)&lds_b[0][0];
    const int NT = K_EMBED / 16;            // 625 tiles
    if (issuer) tdm_load_tile(cm, lds0);    // prefetch tile 0 -> buf0

    for (int t = 0; t < NT; ++t) {
        if (issuer) {
            if (t + 1 < NT) {
                tdm_load_tile(cm + (size_t)(t + 1) * 16,
                              lds0 + (unsigned int)((t + 1) & 1) * (E_DIM * 16 * 4));
                __builtin_amdgcn_s_wait_tensorcnt(1);   // tile t complete
            } else {
                __builtin_amdgcn_s_wait_tensorcnt(0);
            }
        }
        __syncthreads();                    // buf[t&1] visible to all waves

        const float* bt = &lds_b[t & 1][0];
        v8f c0 = {}, c1 = {};               // two independent WMMA chains
        #pragma unroll
        for (int i = 0; i < 32; i += 2) {
            int e0 = 4 * i + 2 * half;
            v2f b0 = { bt[e0 * 16 + l16], bt[(e0 + 1) * 16 + l16] };
            c0 = __builtin_amdgcn_wmma_f32_16x16x4_f32(false, a[i],     false, b0,
                                                       (short)0, c0, false, false);
            int e1 = e0 + 4;
            v2f b1 = { bt[e1 * 16 + l16], bt[(e1 + 1) * 16 + l16] };
            c1 = __builtin_amdgcn_wmma_f32_16x16x4_f32(false, a[i + 1], false, b1,
                                                       (short)0, c1, false, false);
        }

        int   k0 = t * 16;
        float cn = cnorm[k0 + l16];
        int   kk = k0 + l16;
        #pragma unroll
        for (int j = 0; j < 8; ++j) {
            float s = fmaf(-2.0f, c0[j] + c1[j], cn);
            if (s < best_v[j]) { best_v[j] = s; best_i[j] = kk; }
        }
        __syncthreads();                    // done reading buf[t&1]
    }
#else
    for (int k0 = 0; k0 < K_EMBED; k0 += 16) {
        for (int t = threadIdx.x; t < 512; t += 256) {
            int e  = t >> 2;
            int j4 = (t & 3) * 4;
            float4 v = *(const float4*)(cm + (size_t)e * K_EMBED + k0 + j4);
            float* dst = &lds_b[0][0] + (e >> 1) * 32 + (e & 1);
            dst[(j4 + 0) * 2] = v.x;
            dst[(j4 + 1) * 2] = v.y;
            dst[(j4 + 2) * 2] = v.z;
            dst[(j4 + 3) * 2] = v.w;
        }
        __syncthreads();

        v8f c0 = {}, c1 = {};
        #pragma unroll
        for (int i = 0; i < 32; i += 2) {
            int p0 = 2 * i + half;
            v2f b0 = *(const v2f*)(&lds_b[0][0] + p0 * 32 + l16 * 2);
            c0 = __builtin_amdgcn_wmma_f32_16x16x4_f32(false, a[i],     false, b0,
                                                       (short)0, c0, false, false);
            int p1 = p0 + 2;
            v2f b1 = *(const v2f*)(&lds_b[0][0] + p1 * 32 + l16 * 2);
            c1 = __builtin_amdgcn_wmma_f32_16x16x4_f32(false, a[i + 1], false, b1,
                                                       (short)0, c1, false, false);
        }

        float cn = cnorm[k0 + l16];
        int   kk = k0 + l16;
        #pragma unroll
        for (int j = 0; j < 8; ++j) {
            float s = fmaf(-2.0f, c0[j] + c1[j], cn);
            if (s < best_v[j]) { best_v[j] = s; best_i[j] = kk; }
        }
        __syncthreads();
    }
#endif

    // Argmin across the 16 lanes holding the same row (stay within half-wave).
    #pragma unroll
    for (int off = 8; off >= 1; off >>= 1) {
        #pragma unroll
        for (int j = 0; j < 8; ++j) {
            float ov = __shfl_xor(best_v[j], off, 32);
            int   oi = __shfl_xor(best_i[j], off, 32);
            if (ov < best_v[j] || (ov == best_v[j] && oi < best_i[j])) {
                best_v[j] = ov; best_i[j] = oi;
            }
        }
    }
    // C/D layout: lanes 0-15 VGPR j -> M=j ; lanes 16-31 VGPR j -> M=j+8
    if (l16 == 0) {
        #pragma unroll
        for (int j = 0; j < 8; ++j)
            idx[rowBase + half * 8 + j] = best_i[j];
    }
}

// ---------------------------------------------------------------------------
__global__ void k_scatter(const float* __restrict__ x, const int* __restrict__ idx,
                          float* __restrict__ counts, float* __restrict__ sums) {
    int n = blockIdx.x;
    int e = threadIdx.x;
    int k = idx[n];
    atomicAdd(&sums[(size_t)e * K_EMBED + k], x[(size_t)n * E_DIM + e]);
    if (e == 0) atomicAdd(&counts[k], 1.0f);
}

// ---------------------------------------------------------------------------
__global__ void k_ema_size(const float* __restrict__ csize, const float* __restrict__ counts,
                           float* __restrict__ out_size, float* __restrict__ nsample) {
    int k = blockIdx.x * 256 + threadIdx.x;
    float ns = 0.0f;
    if (k < K_EMBED) {
        ns = csize[k] * DECAYF + counts[k] * ONEMD;
        out_size[k] = ns;
    }
    #pragma unroll
    for (int off = 16; off >= 1; off >>= 1) ns += __shfl_down(ns, off, 32);
    if ((threadIdx.x & 31) == 0) atomicAdd(nsample, ns);
}

// ---------------------------------------------------------------------------
__global__ void k_ema_mean(const float* __restrict__ csum, const float* __restrict__ sums,
                           const float* __restrict__ new_size, const float* __restrict__ nsample,
                           float* __restrict__ out_sum, float* __restrict__ out_mean) {
    int i = blockIdx.x * 256 + threadIdx.x;
    if (i >= E_DIM * K_EMBED) return;
    int k = i % K_EMBED;
    float ns_tot = *nsample;
    float nsum = csum[i] * DECAYF + sums[i] * ONEMD;
    out_sum[i] = nsum;
    float smoothed = (new_size[k] + EPSF) * ns_tot / (ns_tot + (float)K_EMBED * EPSF);
    out_mean[i] = nsum / smoothed;
}

// ---------------------------------------------------------------------------
__global__ void k_quantize(const float* __restrict__ x, const int* __restrict__ idx,
                           const float* __restrict__ mean, float* __restrict__ outq,
                           float* __restrict__ loss) {
    __shared__ float red[4];
    int n = blockIdx.x;
    int e = threadIdx.x;
    int k = idx[n];
    float q  = mean[(size_t)e * K_EMBED + k];
    float xi = x[(size_t)n * E_DIM + e];
    outq[(size_t)n * E_DIM + e] = q;       // straight-through output == quantize
    float d = xi - q;
    d = d * d;
    #pragma unroll
    for (int off = 16; off >= 1; off >>= 1) d += __shfl_down(d, off, 32);
    int lane = threadIdx.x & 31, wv = threadIdx.x >> 5;
    if (lane == 0) red[wv] = d;
    __syncthreads();
    if (threadIdx.x == 0) {
        float t = red[0] + red[1] + red[2] + red[3];
        atomicAdd(loss, t * (0.25f / (float)((size_t)N_SAMP * E_DIM)));
    }
}

// ---------------------------------------------------------------------------
extern "C" void kernel_launch(void* const* d_in, const int* in_sizes, int n_in,
                              void* d_out, int out_size, void* d_ws, size_t ws_size,
                              hipStream_t stream) {
    const float* x     = (const float*)d_in[0];   // [16,32,32,128]
    const float* cmean = (const float*)d_in[1];   // [128,10000]
    const float* csum  = (const float*)d_in[2];   // [128,10000]
    const float* csize = (const float*)d_in[3];   // [10000]

    float* out = (float*)d_out;
    float* wsf = (float*)d_ws;

    float* cnorm   = wsf + WS_CNORM;
    int*   idxbuf  = (int*)(wsf + WS_IDX);
    float* counts  = wsf + WS_COUNTS;
    float* sums    = wsf + WS_SUMS;
    float* nsample = wsf + WS_NS;

    float* outq       = out + OUT_Q;
    float* loss       = out + OUT_LOSS;
    float* out_mean   = out + OUT_MEAN;
    float* out_size_p = out + OUT_SIZE;
    float* out_sum    = out + OUT_SUM;

    // 1. zero accumulators (counts, sums, nsample contiguous) + loss
    k_zero<<<(WS_ZERO_N + 255) / 256, 256, 0, stream>>>(counts, WS_ZERO_N, loss);

    // 2. per-code squared norms
    k_colnorm<<<(K_EMBED + 255) / 256, 256, 0, stream>>>(cmean, cnorm);

    // 3. WMMA fp32 GEMM-argmin with TDM double-buffered codebook tiles
    k_argmin<<<N_SAMP / 128, 256, 0, stream>>>(x, cmean, cnorm, idxbuf);

    // 4. segment sums via global f32 atomics
    k_scatter<<<N_SAMP, E_DIM, 0, stream>>>(x, idxbuf, counts, sums);

    // 5. EMA cluster size + total sample count
    k_ema_size<<<(K_EMBED + 255) / 256, 256, 0, stream>>>(csize, counts, out_size_p, nsample);

    // 6. EMA cluster sum + smoothed new means
    k_ema_mean<<<(E_DIM * K_EMBED + 255) / 256, 256, 0, stream>>>(
        csum, sums, out_size_p, nsample, out_sum, out_mean);

    // 7. gather quantized output + commitment loss
    k_quantize<<<N_SAMP, E_DIM, 0, stream>>>(x, idxbuf, out_mean, outq, loss);
}